// HGTLayer_5995774345994
// MI455X (gfx1250) — compile-verified
//
#include <hip/hip_runtime.h>
#include <math.h>

// ---------------------------------------------------------------------------
// HGT layer for MI455X (gfx1250, wave32).
// Matrix ops use native fp32 WMMA: V_WMMA_F32_16X16X4_F32.
// ---------------------------------------------------------------------------

typedef float v2f __attribute__((ext_vector_type(2)));
typedef float v8f __attribute__((ext_vector_type(8)));

#define NNODES 20000
#define NEDGES 320000
#define DMODEL 256
#define NHEAD  8
#define DKH    32

// ---- ordered-uint encoding for float atomicMax ----------------------------
__device__ __forceinline__ unsigned f2ord(float f) {
  unsigned u = __float_as_uint(f);
  return (u & 0x80000000u) ? ~u : (u | 0x80000000u);
}
__device__ __forceinline__ float ord2f(unsigned u) {
  return __uint_as_float((u & 0x80000000u) ? (u ^ 0x80000000u) : ~u);
}

// ---------------------------------------------------------------------------
// Fold per-head relation transform into projection weights:
//   Weff[h*32+e][i] = sum_d rel[h][d][e] * W[h*32+d][i]
// ---------------------------------------------------------------------------
__global__ __launch_bounds__(256) void build_weff(
    const float* __restrict__ W, const float* __restrict__ rel,
    float* __restrict__ Weff) {
  int t = blockIdx.x * blockDim.x + threadIdx.x;   // 65536 threads
  int i  = t & 255;        // input-feature index (coalesced)
  int o  = t >> 8;         // output row = h*32 + e
  int hh = o >> 5, e2 = o & 31;
  float s = 0.f;
  #pragma unroll 8
  for (int d = 0; d < DKH; ++d)
    s += rel[hh * 1024 + d * 32 + e2] * W[(hh * 32 + d) * DMODEL + i];
  Weff[o * DMODEL + i] = s;
}

__global__ __launch_bounds__(256) void build_beff(
    const float* __restrict__ b, const float* __restrict__ rel,
    float* __restrict__ beff) {
  int o  = blockIdx.x * blockDim.x + threadIdx.x;  // 256 threads
  int hh = o >> 5, e2 = o & 31;
  float s = 0.f;
  #pragma unroll 8
  for (int d = 0; d < DKH; ++d)
    s += rel[hh * 1024 + d * 32 + e2] * b[hh * 32 + d];
  beff[o] = s;
}

// ---------------------------------------------------------------------------
// C[M,256] = A[M,256] @ W[256,256]^T + bias   (row-major W: [out,in])
// One wave computes a 16x64 tile: 4 v8f accumulators, K stepped by 4 via
// V_WMMA_F32_16X16X4_F32.  Optional epilogue: out = C*alpha + h*(1-alpha).
//
// Fragment layouts (ISA 7.12.2, 32-bit, wave32):
//   A 16x4 : lane L holds A[M=L&15][K = (L>>4)*2 + {0,1}]   -> v2f
//   B 4x16 : lane L holds B[K = (L>>4)*2 + {0,1}][N=L&15]   -> v2f
//            B[k][n] = W[n][k]  => contiguous float2 from W row n
//   C 16x16: vgpr r, lanes 0-15: M=r, N=lane; lanes 16-31: M=8+r, N=lane-16
// ---------------------------------------------------------------------------
template <bool MIX>
__global__ __launch_bounds__(256) void hgt_gemm_wmma(
    const float* __restrict__ A, const float* __restrict__ W,
    const float* __restrict__ bias, const float* __restrict__ hmix,
    const float* __restrict__ skip, float* __restrict__ C, int M) {
  const int lane  = threadIdx.x & 31;
  const int wave  = threadIdx.x >> 5;
  const int gw    = blockIdx.x * 8 + wave;       // global wave id
  const int mTile = (gw >> 2) * 16;              // 4 N-tiles per M-tile
  const int nTile = (gw & 3) * 64;
  if (mTile >= M) return;

  const int ln = lane & 15;                      // M-row / N-col within tile
  const int ks = (lane >> 4) * 2;                // K sub-offset (0 or 2)

  const float* ap = A + (size_t)(mTile + ln) * DMODEL + ks;
  const float* bp = W + (size_t)(nTile + ln) * DMODEL + ks;

  v8f acc0 = {}, acc1 = {}, acc2 = {}, acc3 = {};

  #pragma unroll 4
  for (int k0 = 0; k0 < DMODEL; k0 += 4) {
    v2f a  = *(const v2f*)(ap + k0);
    v2f b0 = *(const v2f*)(bp + 0 * 16 * DMODEL + k0);
    v2f b1 = *(const v2f*)(bp + 1 * 16 * DMODEL + k0);
    v2f b2 = *(const v2f*)(bp + 2 * 16 * DMODEL + k0);
    v2f b3 = *(const v2f*)(bp + 3 * 16 * DMODEL + k0);
    acc0 = __builtin_amdgcn_wmma_f32_16x16x4_f32(false, a, false, b0, (short)0, acc0, false, false);
    acc1 = __builtin_amdgcn_wmma_f32_16x16x4_f32(false, a, false, b1, (short)0, acc1, false, false);
    acc2 = __builtin_amdgcn_wmma_f32_16x16x4_f32(false, a, false, b2, (short)0, acc2, false, false);
    acc3 = __builtin_amdgcn_wmma_f32_16x16x4_f32(false, a, false, b3, (short)0, acc3, false, false);
  }

  float alpha = 1.f, oneMinus = 0.f;
  if (MIX) {
    alpha    = 1.0f / (1.0f + __expf(-skip[0]));
    oneMinus = 1.0f - alpha;
  }

  const int rbase = (lane >> 4) * 8;             // lanes 16-31 cover M = 8..15
  v8f accs[4] = {acc0, acc1, acc2, acc3};
  #pragma unroll
  for (int i = 0; i < 4; ++i) {
    const int col  = nTile + i * 16 + ln;
    const float bc = bias[col];
    #pragma unroll
    for (int r = 0; r < 8; ++r) {
      const int row = mTile + rbase + r;
      float val = accs[i][r] + bc;
      if (MIX) val = val * alpha + hmix[(size_t)row * DMODEL + col] * oneMinus;
      C[(size_t)row * DMODEL + col] = val;
    }
  }
}

// ---------------------------------------------------------------------------
// Edge scores: sc[e*8+h] = (q[dst[e],h,:] . k[src[e],h,:]) * rel_pri[h]/sqrt(32)
// ---------------------------------------------------------------------------
__global__ __launch_bounds__(256) void hgt_edge_score(
    const float* __restrict__ q, const float* __restrict__ k,
    const int* __restrict__ src, const int* __restrict__ dst,
    const float* __restrict__ rel_pri, float* __restrict__ sc, int EH) {
  int t = blockIdx.x * blockDim.x + threadIdx.x;
  if (t >= EH) return;
  int e = t >> 3, hh = t & 7;
  const float4* qp = (const float4*)(q + (size_t)dst[e] * DMODEL + hh * DKH);
  const float4* kp = (const float4*)(k + (size_t)src[e] * DMODEL + hh * DKH);
  float s = 0.f;
  #pragma unroll
  for (int i = 0; i < 8; ++i) {
    float4 a = qp[i], b = kp[i];
    s += a.x * b.x + a.y * b.y + a.z * b.z + a.w * b.w;
  }
  sc[t] = s * rel_pri[hh] * 0.17677669529663687f;   // 1/sqrt(32)
}

__global__ __launch_bounds__(256) void fill_u32(unsigned* __restrict__ p,
                                                unsigned val, int n) {
  int t = blockIdx.x * blockDim.x + threadIdx.x;
  if (t < n) p[t] = val;
}

__global__ __launch_bounds__(256) void hgt_segmax(
    const float* __restrict__ sc, const int* __restrict__ dst,
    unsigned* __restrict__ m, int EH) {
  int t = blockIdx.x * blockDim.x + threadIdx.x;
  if (t >= EH) return;
  int e = t >> 3, hh = t & 7;
  atomicMax(m + (size_t)dst[e] * NHEAD + hh, f2ord(sc[t]));
}

__global__ __launch_bounds__(256) void hgt_expz(
    float* __restrict__ sc, const int* __restrict__ dst,
    const unsigned* __restrict__ m, float* __restrict__ z, int EH) {
  int t = blockIdx.x * blockDim.x + threadIdx.x;
  if (t >= EH) return;
  int e = t >> 3, hh = t & 7;
  float mx = ord2f(m[(size_t)dst[e] * NHEAD + hh]);
  float ex = __expf(sc[t] - mx);
  sc[t] = ex;
  atomicAdd(z + (size_t)dst[e] * NHEAD + hh, ex);
}

// ---------------------------------------------------------------------------
// agg[dst,h,:] += (ex/z[dst,h]) * v[src,h,:]
// ---------------------------------------------------------------------------
__global__ __launch_bounds__(256) void hgt_aggregate(
    const float* __restrict__ ex, const float* __restrict__ z,
    const float* __restrict__ v, const int* __restrict__ src,
    const int* __restrict__ dst, float* __restrict__ agg, int EH) {
  int t = blockIdx.x * blockDim.x + threadIdx.x;
  if (t >= EH) return;
  int e = t >> 3, hh = t & 7;
  int d0 = dst[e];
  float a = ex[t] / z[(size_t)d0 * NHEAD + hh];
  const float* vp = v + (size_t)src[e] * DMODEL + hh * DKH;
  float* op = agg + (size_t)d0 * DMODEL + hh * DKH;
  #pragma unroll
  for (int i = 0; i < DKH; i += 4) {
    float4 vv = *(const float4*)(vp + i);
    atomicAdd(op + i + 0, a * vv.x);
    atomicAdd(op + i + 1, a * vv.y);
    atomicAdd(op + i + 2, a * vv.z);
    atomicAdd(op + i + 3, a * vv.w);
  }
}

// ---------------------------------------------------------------------------
extern "C" void kernel_launch(void* const* d_in, const int* in_sizes, int n_in,
                              void* d_out, int out_size, void* d_ws,
                              size_t ws_size, hipStream_t stream) {
  const float* h       = (const float*)d_in[0];
  const int*   src     = (const int*)d_in[1];
  const int*   dst     = (const int*)d_in[2];
  const float* Wk      = (const float*)d_in[3];
  const float* bk      = (const float*)d_in[4];
  const float* Wq      = (const float*)d_in[5];
  const float* bq      = (const float*)d_in[6];
  const float* Wv      = (const float*)d_in[7];
  const float* bv      = (const float*)d_in[8];
  const float* Wa      = (const float*)d_in[9];
  const float* ba      = (const float*)d_in[10];
  const float* rel_att = (const float*)d_in[11];
  const float* rel_msg = (const float*)d_in[12];
  const float* rel_pri = (const float*)d_in[13];
  const float* skip    = (const float*)d_in[14];
  float* out = (float*)d_out;

  const int EH = NEDGES * NHEAD;                 // 2,560,000

  // workspace layout (floats); buffers reused: q->v, k->agg
  float*    ws  = (float*)d_ws;
  float*    q   = ws;                                   // [N,256] q, later v
  float*    kv  = q  + (size_t)NNODES * DMODEL;         // [N,256] k, later agg
  float*    sc  = kv + (size_t)NNODES * DMODEL;         // [E,8] scores / exp
  unsigned* m   = (unsigned*)(sc + (size_t)EH);         // [N,8] ordered max
  float*    z   = (float*)(m + (size_t)NNODES * NHEAD); // [N,8] softmax denom
  float*    WeK = z + (size_t)NNODES * NHEAD;           // [256,256]
  float*    WeV = WeK + DMODEL * DMODEL;                // [256,256]
  float*    beK = WeV + DMODEL * DMODEL;                // [256]
  float*    beV = beK + DMODEL;                         // [256]

  // 1) fold relation transforms into projection weights/biases
  build_weff<<<256, 256, 0, stream>>>(Wk, rel_att, WeK);
  build_weff<<<256, 256, 0, stream>>>(Wv, rel_msg, WeV);
  build_beff<<<1, 256, 0, stream>>>(bk, rel_att, beK);
  build_beff<<<1, 256, 0, stream>>>(bv, rel_msg, beV);

  // 2) q and k projections (WMMA f32).  20000/16 * 4 ntiles / 8 waves = 625
  dim3 gg(625), bb(256);
  hgt_gemm_wmma<false><<<gg, bb, 0, stream>>>(h, Wq,  bq,  nullptr, nullptr, q,  NNODES);
  hgt_gemm_wmma<false><<<gg, bb, 0, stream>>>(h, WeK, beK, nullptr, nullptr, kv, NNODES);

  // 3) edge scores (consumes q and kv)
  hgt_edge_score<<<(EH + 255) / 256, 256, 0, stream>>>(q, kv, src, dst, rel_pri, sc, EH);

  // 4) v projection — overwrites q (dead after scores)
  hgt_gemm_wmma<false><<<gg, bb, 0, stream>>>(h, WeV, beV, nullptr, nullptr, q, NNODES);

  // 5) init segment buffers (every call: workspace not re-poisoned but we accumulate)
  fill_u32<<<(NNODES * NHEAD + 255) / 256, 256, 0, stream>>>(m, 0x007FFFFFu,
                                                             NNODES * NHEAD); // ord(-inf)
  hipMemsetAsync(z, 0, sizeof(float) * (size_t)NNODES * NHEAD, stream);
  hipMemsetAsync(kv, 0, sizeof(float) * (size_t)NNODES * DMODEL, stream);

  // 6) segment softmax over dst
  hgt_segmax<<<(EH + 255) / 256, 256, 0, stream>>>(sc, dst, m, EH);
  hgt_expz<<<(EH + 255) / 256, 256, 0, stream>>>(sc, dst, m, z, EH);

  // 7) attention-weighted aggregation into kv (= agg)
  hgt_aggregate<<<(EH + 255) / 256, 256, 0, stream>>>(sc, z, q, src, dst, kv, EH);

  // 8) output GEMM + skip-mix epilogue
  hgt_gemm_wmma<true><<<gg, bb, 0, stream>>>(kv, Wa, ba, h, skip, out, NNODES);
}